// FixedPositionRoPE_18519898980567
// MI455X (gfx1250) — compile-verified
//
#include <hip/hip_runtime.h>

// RoPE with fixed positions {0,1} on x:(4,4096,16,128) fp32.
// Bandwidth-bound streaming kernel for MI455X (23.3 TB/s HBM, 192MB L2):
//   - 268 MB touched once -> non-temporal b128 loads/stores
//   - 4 heads per thread: cos/sin (12 TRANS ops) computed once, reused 4x
//   - block = 256 threads = 4 tokens x (16 quad-groups x 4 head-groups);
//     each wave32 stays within one token -> pos-dependent math is uniform
//   - per vmem instruction a wave touches 2x256B contiguous segments

typedef float v4f __attribute__((ext_vector_type(4)));

#define ROPE_DIM    128
#define ROPE_HALF   64
#define ROPE_HEADS  16
#define TOKENS_PER_BLOCK 4

__global__ __launch_bounds__(256)
void FixedPositionRoPE_18519898980567_kernel(const float* __restrict__ x,
                                             const int*   __restrict__ pos_ids,
                                             float*       __restrict__ out,
                                             int n_tokens)
{
    const int tid = threadIdx.x;
    const int tok = blockIdx.x * TOKENS_PER_BLOCK + (tid >> 6);
    if (tok >= n_tokens) return;

    const int sub = tid & 63;
    const int qg  = sub & 15;          // column quad group: cols [4*qg, 4*qg+3]
    const int hg  = sub >> 4;          // head group 0..3 -> heads 4*hg .. 4*hg+3

    // Wave-uniform position (whole wave serves one token).
    int p = pos_ids[tok];
    p = (p < 0) ? 0 : ((p > 1) ? 1 : p);     // jnp.clip(position_ids, 0, 1)
    const float pf = (float)p;

    // inv_freq[c] = 10000^(-c/64) = exp2(c * (-log2(10000)/64))
    // Computed ONCE per thread, reused for 4 head rows (32 output floats).
    const float kexp = -0.20762050593045951f; // log2(10000)/64
    v4f cs, sn;
    #pragma unroll
    for (int i = 0; i < 4; ++i) {
        const float c   = (float)((qg << 2) + i);
        const float ang = pf * exp2f(c * kexp);  // pos==0 -> ang=0 -> cos=1,sin=0
        float s, co;
        __sincosf(ang, &s, &co);                 // v_sin_f32 / v_cos_f32 TRANS
        cs[i] = co;
        sn[i] = s;
    }

    const long long tok_base = (long long)tok * (ROPE_HEADS * ROPE_DIM);

    #pragma unroll
    for (int h = 0; h < 4; ++h) {
        const int head = (hg << 2) + h;
        const long long base = tok_base + head * ROPE_DIM + (qg << 2);

        // 128-bit non-temporal loads of the real/imag halves (16B aligned).
        const v4f xr = __builtin_nontemporal_load((const v4f*)(x + base));
        const v4f xi = __builtin_nontemporal_load((const v4f*)(x + base + ROPE_HALF));

        const v4f o_re = xr * cs - xi * sn;
        const v4f o_im = xi * cs + xr * sn;

        // 128-bit non-temporal stores: streamed once, keep L2 clean.
        __builtin_nontemporal_store(o_re, (v4f*)(out + base));
        __builtin_nontemporal_store(o_im, (v4f*)(out + base + ROPE_HALF));
    }
}

extern "C" void kernel_launch(void* const* d_in, const int* in_sizes, int n_in,
                              void* d_out, int out_size, void* d_ws, size_t ws_size,
                              hipStream_t stream)
{
    const float* x       = (const float*)d_in[0];   // (4,4096,16,128) fp32
    const int*   pos_ids = (const int*)  d_in[1];   // (4,4096) int
    float*       out     = (float*)d_out;

    const int n_tokens = in_sizes[1];               // 4*4096 = 16384
    const int n_blocks = (n_tokens + TOKENS_PER_BLOCK - 1) / TOKENS_PER_BLOCK;

    dim3 grid(n_blocks);
    dim3 block(256);   // 4 tokens x 64 threads (16 quad-groups x 4 head-groups)
    FixedPositionRoPE_18519898980567_kernel<<<grid, block, 0, stream>>>(
        x, pos_ids, out, n_tokens);
}